// AttentionAggregator_15934328668386
// MI455X (gfx1250) — compile-verified
//
#include <hip/hip_runtime.h>

// ---------------------------------------------------------------------------
// AttentionAggregator on MI455X (gfx1250).
//
// The reference collapses algebraically: softmax over rows makes every column
// of `a` sum to 1, so  agg[b,n,d] = sum_{p=start..end} embeddings[b,p,d].
// W and b cancel entirely. The kernel is therefore a masked segment-sum:
// ~50 MFLOP, ~60 MB unique traffic -> pure HBM/L2 bandwidth bound (~3us).
//
// We run the sum through the CDNA5 matrix pipe (free at this arithmetic
// intensity): per span, C[16x16] += A[16x4] x B[4x16] with
//   A[m][k] = mask(k)  (1.0f / 0.0f, broadcast down rows -> exact f32 math)
//   B[k][d] = emb[pos(k)][dtile+d]
// accumulated over 4 K-chunks (K total = LMAX = 16), D tiled by 16.
// Uses V_WMMA_F32_16X16X4_F32 -> f32 products by {0,1} and f32 accumulation,
// identical precision to a scalar f32 sum.
// ---------------------------------------------------------------------------

typedef __attribute__((ext_vector_type(2))) float v2f;
typedef __attribute__((ext_vector_type(8))) float v8f;

#define BB   32
#define SS   512
#define DD   768
#define NN   128
#define LMAX 16

__global__ __launch_bounds__(256)
void span_sum_wmma(const float* __restrict__ emb,    // [B,S,D] f32
                   const int*   __restrict__ spans,  // [B,N,2] i32 (start, end incl.)
                   float*       __restrict__ out)    // [B,N,D] f32
{
    const int lane = threadIdx.x & 31;
    const int wave = threadIdx.x >> 5;                      // 8 waves / block
    const int span = blockIdx.x * (blockDim.x >> 5) + wave; // one wave per span
    if (span >= BB * NN) return;                            // wave-uniform guard

    const int b     = span / NN;
    const int start = spans[(span << 1) + 0];
    const int end   = spans[(span << 1) + 1];               // inclusive

    const int half = lane >> 4;   // which 16-lane half of the wave
    const int col  = lane & 15;   // N (column) index inside the 16-wide D tile

    const float* __restrict__ embB = emb + (size_t)b * SS * DD;
    float*       __restrict__ outp = out + (size_t)span * DD;

    for (int dt = 0; dt < DD; dt += 16) {
        v8f acc = {};
        #pragma unroll
        for (int kc = 0; kc < 4; ++kc) {
            // ISA 7.12.2 layouts for 16x16x4 f32:
            //   A: VGPR0 holds K = 0(+2*half), VGPR1 holds K = 1(+2*half)
            //   B: same K striping; columns N=0..15 across lanes of each half
            const int k0 = kc * 4 + 2 * half;
            const int k1 = k0 + 1;
            const int p0r = start + k0;
            const int p1r = start + k1;
            const float m0 = (p0r <= end) ? 1.0f : 0.0f;
            const float m1 = (p1r <= end) ? 1.0f : 0.0f;
            const int p0 = (p0r < SS) ? p0r : (SS - 1);   // matches jnp.clip
            const int p1 = (p1r < SS) ? p1r : (SS - 1);

            v2f a;  a.x  = m0;  a.y  = m1;                // mask, broadcast over M
            v2f bm; bm.x = embB[(size_t)p0 * DD + dt + col];
                    bm.y = embB[(size_t)p1 * DD + dt + col];

            // 8 args: (neg_a, A, neg_b, B, c_mod, C, reuse_a, reuse_b)
            acc = __builtin_amdgcn_wmma_f32_16x16x4_f32(
                    false, a, false, bm, (short)0, acc, false, false);
        }
        // Every row of C is identical; row M=0 lives in acc[0] of lanes 0-15.
        if (half == 0) outp[dt + col] = acc[0];
    }
}

extern "C" void kernel_launch(void* const* d_in, const int* in_sizes, int n_in,
                              void* d_out, int out_size, void* d_ws, size_t ws_size,
                              hipStream_t stream) {
    // setup_inputs order: embeddings, W, b, spans.  W and b cancel (see header).
    const float* emb   = (const float*)d_in[0];
    const int*   spans = (const int*)d_in[3];
    float*       out   = (float*)d_out;

    const int total_spans = BB * NN;            // 4096 waves, one per span
    const int waves_per_block = 256 / 32;       // 8
    dim3 grid((total_spans + waves_per_block - 1) / waves_per_block);
    dim3 block(256);
    hipLaunchKernelGGL(span_sum_wmma, grid, block, 0, stream, emb, spans, out);
}